// ScorePredictor_33122787786912
// MI455X (gfx1250) — compile-verified
//
#include <hip/hip_runtime.h>
#include <math.h>

typedef __attribute__((ext_vector_type(2))) float v2f;
typedef __attribute__((ext_vector_type(8))) float v8f;

#define DFEAT 128

// Phase 1: per-node projections ps[n] = x[n].w_s, pd[n] = x[n].w_d
// One wave32 handles a 16-row tile using V_WMMA_F32_16X16X4_F32, K-loop of 32.
__global__ __launch_bounds__(128) void node_proj_wmma(
    const float* __restrict__ x,
    const float* __restrict__ W,   // [0..127]=w_s, [128..255]=w_d
    float* __restrict__ ps,
    float* __restrict__ pd,
    int n_nodes) {
  const int lane = threadIdx.x & 31;
  const int wave = threadIdx.x >> 5;
  const int tile = blockIdx.x * (blockDim.x >> 5) + wave;
  const long base = (long)tile * 16;
  if (base >= n_nodes) return;              // uniform per-wave: EXEC stays full

  const int col  = lane & 15;               // M (for A) / N (for B,C)
  const int half = lane >> 4;               // selects K sub-pair per ISA layout

  // A: lane reads row (base+col), K pair = 4s + 2*half + {0,1}
  long rowA = base + col;
  if (rowA >= n_nodes) rowA = n_nodes - 1;  // clamp, keep EXEC full
  const float* aptr = x + rowA * DFEAT + half * 2;

  // B: column 0 = w_s, column 1 = w_d, columns 2..15 = 0
  const int wsel = (col < 2) ? col : 0;
  const float* bptr = W + wsel * DFEAT + half * 2;
  const bool bzero = (col >= 2);

  v8f c = {};
#pragma unroll 8
  for (int s = 0; s < DFEAT / 4; ++s) {
    v2f a  = *(const v2f*)(aptr + s * 4);
    v2f bv = *(const v2f*)(bptr + s * 4);
    if (bzero) { bv.x = 0.0f; bv.y = 0.0f; }
    // 8 args: (neg_a, A, neg_b, B, c_mod, C, reuse_a, reuse_b)
    c = __builtin_amdgcn_wmma_f32_16x16x4_f32(false, a, false, bv,
                                              (short)0, c, false, false);
  }

  // C layout: VGPR r -> lanes 0..15: M=r, N=lane ; lanes 16..31: M=r+8, N=lane-16
  // Lane (col==0, half) holds ps[base+half*8 .. +7] in c[0..7]; col==1 -> pd.
  float4 lo = make_float4(c[0], c[1], c[2], c[3]);
  float4 hi = make_float4(c[4], c[5], c[6], c[7]);
  float* outp = (col == 0) ? ps : pd;
  const long rbase = base + half * 8;

  if (base + 16 <= (long)n_nodes) {         // uniform: full tile, wide stores
    if (col < 2) {
      *(float4*)(outp + rbase)     = lo;
      *(float4*)(outp + rbase + 4) = hi;
    }
  } else {                                  // tail tile (unused for N%16==0)
    if (col < 2) {
#pragma unroll
      for (int r = 0; r < 8; ++r) {
        long row = rbase + r;
        if (row < n_nodes) outp[row] = c[r];
      }
    }
  }
}

__device__ __forceinline__ float sigmoidf(float v) {
  return 1.0f / (1.0f + expf(-v));
}

// Phase 2: out[e] = sigmoid(ps[src[e]] + pd[dst[e]] + b), 4 edges per thread
__global__ __launch_bounds__(256) void edge_score(
    const int* __restrict__ src,
    const int* __restrict__ dst,
    const float* __restrict__ ps,
    const float* __restrict__ pd,
    const float* __restrict__ b,
    float* __restrict__ out,
    int n_edges) {
  const int i  = blockIdx.x * blockDim.x + threadIdx.x;
  const int n4 = n_edges >> 2;
  const float bb = b[0];
  if (i < n4) {
    int4 s = ((const int4*)src)[i];
    int4 d = ((const int4*)dst)[i];
    float4 o;
    o.x = sigmoidf(ps[s.x] + pd[d.x] + bb);
    o.y = sigmoidf(ps[s.y] + pd[d.y] + bb);
    o.z = sigmoidf(ps[s.z] + pd[d.z] + bb);
    o.w = sigmoidf(ps[s.w] + pd[d.w] + bb);
    ((float4*)out)[i] = o;
  }
  // tail (n_edges % 4 != 0): first few threads of grid handle leftovers
  const int rem = n_edges & 3;
  if (i < rem) {
    const int e = n4 * 4 + i;
    out[e] = sigmoidf(ps[src[e]] + pd[dst[e]] + bb);
  }
}

extern "C" void kernel_launch(void* const* d_in, const int* in_sizes, int n_in,
                              void* d_out, int out_size, void* d_ws, size_t ws_size,
                              hipStream_t stream) {
  const float* x   = (const float*)d_in[0];
  const int*   src = (const int*)d_in[1];
  const int*   dst = (const int*)d_in[2];
  const float* W   = (const float*)d_in[3];
  const float* b   = (const float*)d_in[4];
  float* out = (float*)d_out;

  const int n_nodes = in_sizes[0] / DFEAT;
  const int n_edges = in_sizes[1];

  float* ps = (float*)d_ws;
  float* pd = ps + n_nodes;

  const int tiles = (n_nodes + 15) / 16;
  const int waves_per_block = 4;  // 128 threads = 4 wave32
  const int blocks1 = (tiles + waves_per_block - 1) / waves_per_block;
  node_proj_wmma<<<blocks1, waves_per_block * 32, 0, stream>>>(x, W, ps, pd, n_nodes);

  const int work2   = (n_edges + 3) / 4;
  const int blocks2 = (work2 + 255) / 256;
  edge_score<<<blocks2, 256, 0, stream>>>(src, dst, ps, pd, b, out, n_edges);
}